// LoRALinear_15814069584456
// MI455X (gfx1250) — compile-verified
//
#include <hip/hip_runtime.h>
#include <hip/hip_bf16.h>

// ---------------------------------------------------------------------------
// LoRA linear: out = x @ (W^T + (ALPHA/RANK) * B@A) + bias
//   x   : [65536, 1024] f32      W : [1024(out), 1024(in)] f32
//   bias: [1024] f32             B : [1024(in), 16] f32     A : [16, 1024] f32
// Strategy:
//   (1) weff_kernel: Weff[k][n] = W[n][k] + scale * sum_r B[k][r]*A[r][n]
//       (4 MB in d_ws; transposes W once so GEMM reads are coalesced;
//        Weff stays resident in the 192 MB L2 for the whole GEMM)
//   (2) lora_gemm_kernel: f32 GEMM on V_WMMA_F32_16X16X4_F32 (exact f32),
//       128x128 block tile, 8 wave32s (2x4), 64x32 per wave, BK=16,
//       double-buffered LDS fed by GLOBAL_LOAD_ASYNC_TO_LDS_B128 (ASYNCcnt),
//       conflict-free fragment gathers via ds_load_2addr.
// ---------------------------------------------------------------------------

typedef __attribute__((ext_vector_type(2))) float v2f;
typedef __attribute__((ext_vector_type(8))) float v8f;

#define LORA_RANK 16
#define LORA_ALPHA 32.0f
#define IN_F 1024
#define OUT_F 1024

#define BM 128
#define BN 128
#define BK 16
#define A_PITCH 20   // 128 rows x 20 f32: row*20 banks distinct for lanes 0..15
#define B_PITCH 136  // 16 rows x 136 f32: +2 rows => +16 banks, halves disjoint

// ---------------------------------------------------------------------------
// Kernel 1: effective weight (also transposes W: Weff is [K=in][N=out])
// ---------------------------------------------------------------------------
__global__ __launch_bounds__(256) void weff_kernel(
    const float* __restrict__ W, const float* __restrict__ B,
    const float* __restrict__ A, float* __restrict__ Weff) {
  const int idx = blockIdx.x * 256 + threadIdx.x;  // 1M outputs
  const int k = idx >> 10;
  const int n = idx & 1023;
  float acc = 0.0f;
#pragma unroll
  for (int r = 0; r < LORA_RANK; ++r)
    acc = fmaf(B[k * LORA_RANK + r], A[r * OUT_F + n], acc);
  Weff[idx] = W[n * IN_F + k] + (LORA_ALPHA / (float)LORA_RANK) * acc;
}

// ---------------------------------------------------------------------------
// Async global->LDS B128 (gfx1250). GVS mode: SGPR64 base + VGPR i32 offset.
// Per ISA pseudocode, INST_OFFSET is added to BOTH the LDS and global
// addresses, so the same lds/global operands serve offset:0 and offset:16.
// Tracked by ASYNCcnt; complete with s_wait_asynccnt.
// ---------------------------------------------------------------------------
template <int IMM>
__device__ __forceinline__ void async_ld_b128(unsigned lds_off,
                                              const float* base,
                                              unsigned byte_off) {
  asm volatile("global_load_async_to_lds_b128 %0, %1, %2 offset:%3"
               :
               : "v"(lds_off), "v"(byte_off), "s"(base), "n"(IMM)
               : "memory");
}

__device__ __forceinline__ void wait_async0() {
  asm volatile("s_wait_asynccnt 0x0" ::: "memory");
}

// ---------------------------------------------------------------------------
// Kernel 2: WMMA f32 GEMM  out[m][n] = sum_k x[m][k]*Weff[k][n] + bias[n]
// ---------------------------------------------------------------------------
__global__ __launch_bounds__(256) void lora_gemm_kernel(
    const float* __restrict__ x, const float* __restrict__ Weff,
    const float* __restrict__ bias, float* __restrict__ out) {
  __shared__ float As[2][BM][A_PITCH];   // x tile, [m][k]
  __shared__ float Bs[2][BK][B_PITCH];   // Weff tile, [k][n]

  const int nBase = blockIdx.x * BN;
  const int mBase = blockIdx.y * BM;
  const int tid   = threadIdx.x;
  const int wave  = tid >> 5;
  const int lane  = tid & 31;
  const int l16   = lane & 15;
  const int lhalf = lane >> 4;

  const int mWave = (wave & 1) * 64;   // 2 waves along M
  const int nWave = (wave >> 1) * 32;  // 4 waves along N

  // global->LDS staging roles (all transfers are B128)
  const int arow = tid >> 2;        // 0..63  (x rows, 2 passes of 64)
  const int acol = (tid & 3) * 4;   // 0,4,8,12 (k within tile)
  const int bk   = tid >> 4;        // 0..15  (k row of Weff tile)
  const int bn   = (tid & 15) * 8;  // 0..120 (n within tile, 2x b128)

  const float* xg = x + (size_t)mBase * IN_F;   // uniform SGPR base
  const float* wg = Weff + nBase;               // uniform SGPR base

  // LDS byte offsets (low 32 bits of the generic pointer == LDS offset)
  const unsigned ldsA = (unsigned)(uintptr_t)&As[0][arow][acol];
  const unsigned ldsB = (unsigned)(uintptr_t)&Bs[0][bk][bn];
  const unsigned A_BUF = BM * A_PITCH * 4;  // 10240 B per buffer
  const unsigned B_BUF = BK * B_PITCH * 4;  // 8704 B per buffer
  const unsigned A_P1  = 64 * A_PITCH * 4;  // second 64-row pass

  auto issue_tile = [&](int buf, int k0) {
    const unsigned la  = ldsA + (unsigned)buf * A_BUF;
    const unsigned lb  = ldsB + (unsigned)buf * B_BUF;
    const unsigned ga0 = (unsigned)((arow * IN_F + k0 + acol) * 4);
    const unsigned ga1 = (unsigned)(((64 + arow) * IN_F + k0 + acol) * 4);
    const unsigned gb0 = (unsigned)(((k0 + bk) * OUT_F + bn) * 4);
    async_ld_b128<0>(la, xg, ga0);           // x rows 0..63
    async_ld_b128<0>(la + A_P1, xg, ga1);    // x rows 64..127
    async_ld_b128<0>(lb, wg, gb0);           // Weff n8..n8+3
    async_ld_b128<16>(lb, wg, gb0);          // Weff n8+4..n8+7
  };

  v8f acc[4][2];
#pragma unroll
  for (int mt = 0; mt < 4; ++mt)
#pragma unroll
    for (int nt = 0; nt < 2; ++nt) acc[mt][nt] = (v8f)0.0f;

  issue_tile(0, 0);
  wait_async0();
  __syncthreads();

  const int KTILES = IN_F / BK;  // 64
  for (int t = 0; t < KTILES; ++t) {
    const int cur = t & 1;
    if (t + 1 < KTILES) {
      issue_tile(cur ^ 1, (t + 1) * BK);  // DMA overlaps the WMMA stream
      if (t + 2 < KTILES)  // gfx1250 global_prefetch_b8: warm L2 for t+2
        __builtin_prefetch(xg + (size_t)arow * IN_F + (t + 2) * BK + acol, 0, 0);
    }

#pragma unroll
    for (int kk = 0; kk < BK; kk += 4) {
      // A fragments: 16x4 f32, lane L holds A[L&15][2*(L>>4) .. +1]
      v2f af[4];
#pragma unroll
      for (int mt = 0; mt < 4; ++mt) {
        const int row = mWave + mt * 16 + l16;
        af[mt] = *(const v2f*)&As[cur][row][kk + 2 * lhalf];
      }
      // B fragments: 4x16 f32, lane L holds B[2*(L>>4)][L&15], B[2*(L>>4)+1][L&15]
      v2f bf[2];
#pragma unroll
      for (int nt = 0; nt < 2; ++nt) {
        const int col = nWave + nt * 16 + l16;
        v2f b;
        b.x = Bs[cur][kk + 2 * lhalf + 0][col];
        b.y = Bs[cur][kk + 2 * lhalf + 1][col];
        bf[nt] = b;
      }
#pragma unroll
      for (int mt = 0; mt < 4; ++mt)
#pragma unroll
        for (int nt = 0; nt < 2; ++nt)
          acc[mt][nt] = __builtin_amdgcn_wmma_f32_16x16x4_f32(
              false, af[mt], false, bf[nt], (short)0, acc[mt][nt],
              false, false);
    }

    if (t + 1 < KTILES) {
      wait_async0();     // ASYNCcnt not covered by __syncthreads' waits
      __syncthreads();
    }
  }

  // Epilogue: C/D layout -> VGPR v holds M = v + 8*(lane>=16), N = lane&15
  const float b0 = bias[nBase + nWave + l16];
  const float b1 = bias[nBase + nWave + 16 + l16];
#pragma unroll
  for (int mt = 0; mt < 4; ++mt) {
    const int rbase = mBase + mWave + mt * 16 + 8 * lhalf;
#pragma unroll
    for (int nt = 0; nt < 2; ++nt) {
      const int col = nBase + nWave + nt * 16 + l16;
      const float bb = nt ? b1 : b0;
#pragma unroll
      for (int v = 0; v < 8; ++v)
        out[(size_t)(rbase + v) * OUT_F + col] = acc[mt][nt][v] + bb;
    }
  }
}

// ---------------------------------------------------------------------------
extern "C" void kernel_launch(void* const* d_in, const int* in_sizes, int n_in,
                              void* d_out, int out_size, void* d_ws, size_t ws_size,
                              hipStream_t stream) {
  const float* x    = (const float*)d_in[0];
  const float* W    = (const float*)d_in[1];
  const float* bias = (const float*)d_in[2];
  const float* B    = (const float*)d_in[3];
  const float* A    = (const float*)d_in[4];
  float* out  = (float*)d_out;
  float* Weff = (float*)d_ws;  // needs IN_F*OUT_F*4 = 4 MB of scratch

  const int mTotal = in_sizes[0] / IN_F;  // 8*8192 = 65536

  weff_kernel<<<(IN_F * OUT_F) / 256, 256, 0, stream>>>(W, B, A, Weff);

  dim3 grid(OUT_F / BN, mTotal / BM);
  lora_gemm_kernel<<<grid, 256, 0, stream>>>(x, Weff, bias, out);
}